// FurthestPointSampler_9242769621828
// MI455X (gfx1250) — compile-verified
//
#include <hip/hip_runtime.h>

#define TPB 1024
#define INF_F 1.0e10f
#define MAG_EPS 1.0e-3f

// Stage one point's 12 bytes (x,y,z) from global to LDS via the CDNA5 async
// data path. The instruction offset is applied to BOTH the LDS and global
// addresses (CDNA5 ISA), so one VGPR pair covers all 3 dwords.
// s_wait_dscnt 0 first guarantees our previous ds_loads from the buffer we are
// about to overwrite have fully completed before the DMA write can land.
__device__ __forceinline__ void stage3(const float* __restrict__ src,
                                       unsigned ldsOff, unsigned gOff) {
  asm volatile(
      "s_wait_dscnt 0x0\n\t"
      "global_load_async_to_lds_b32 %0, %1, %2 offset:0\n\t"
      "global_load_async_to_lds_b32 %0, %1, %2 offset:4\n\t"
      "global_load_async_to_lds_b32 %0, %1, %2 offset:8"
      :
      : "v"(ldsOff), "v"(gOff), "s"(src)
      : "memory");
}

__global__ __launch_bounds__(TPB, 1) void fps_kernel(
    const float* __restrict__ x,   // (B, N, 3)
    float* __restrict__ y,         // (B, M, 3)
    float* __restrict__ temp,      // (B, N) workspace
    int N, int M) {
  __shared__ float tile[2][3 * TPB];   // double-buffered coord tiles (24 KB)
  __shared__ float sVal[TPB / 32];
  __shared__ int   sIdx[TPB / 32];
  __shared__ float sPoint[3];

  const int b    = blockIdx.x;
  const int tid  = threadIdx.x;
  const int lane = tid & 31;           // wave32
  const int wid  = tid >> 5;
  const int nchunk = N / TPB;

  const float* xb = x + (size_t)b * N * 3;
  float*       tb = temp + (size_t)b * N;
  float*       yb = y + (size_t)b * M * 3;

  // ---- init temp: fold the magnitude mask in. masked -> -1 (matches the
  // reference's where(mask, temp, -1) candidate), valid -> INF. Each thread
  // owns indices {c*TPB + tid}, same ownership as the main loop -> no barrier.
  for (int c = 0; c < nchunk; ++c) {
    const int i = c * TPB + tid;
    const float xx = xb[(size_t)i * 3 + 0];
    const float yy = xb[(size_t)i * 3 + 1];
    const float zz = xb[(size_t)i * 3 + 2];
    const float mag = xx * xx + yy * yy + zz * zz;
    tb[i] = (mag > MAG_EPS) ? INF_F : -1.0f;
  }

  // first pick is always index 0
  float px = xb[0], py = xb[1], pz = xb[2];
  if (tid == 0) { yb[0] = px; yb[1] = py; yb[2] = pz; }

  const unsigned gOff = (unsigned)tid * 12u;
  const unsigned lds0 = (unsigned)(size_t)&tile[0][tid * 3];
  const unsigned lds1 = (unsigned)(size_t)&tile[1][tid * 3];

  for (int s = 1; s < M; ++s) {
    float best = -3.0e38f;
    int bestIdx = 0;

    // prologue: stage chunk 0 into buffer 0
    stage3(xb, lds0, gOff);

    for (int c = 0; c < nchunk; ++c) {
      if (c + 1 < nchunk) {
        // stage next chunk into the other buffer, then wait until the
        // current chunk's 3 async loads (issued earlier, complete in order)
        // are done: ASYNCcnt <= 3 leaves only the next tile in flight.
        stage3(xb + (size_t)(c + 1) * 3 * TPB,
               ((c + 1) & 1) ? lds1 : lds0, gOff);
        asm volatile("s_wait_asynccnt 0x3" ::: "memory");
      } else {
        asm volatile("s_wait_asynccnt 0x0" ::: "memory");
      }

      const float* tp = &tile[c & 1][tid * 3];
      const float xx = tp[0];
      const float yy = tp[1];
      const float zz = tp[2];
      const int i = c * TPB + tid;

      float t = tb[i];
      if (t >= 0.0f) {                 // valid (unmasked) point
        const float dx = xx - px, dy = yy - py, dz = zz - pz;
        const float d = dx * dx + dy * dy + dz * dz;
        if (d < t) { t = d; tb[i] = t; }  // running min; rare store late on
      }
      // ascending i per thread + strict '>' == first-occurrence argmax
      if (t > best) { best = t; bestIdx = i; }
    }

    // ---- wave32 shuffle reduction (max, lower index wins ties)
#pragma unroll
    for (int off = 16; off > 0; off >>= 1) {
      const float ov = __shfl_down(best, off, 32);
      const int   oi = __shfl_down(bestIdx, off, 32);
      if (ov > best || (ov == best && oi < bestIdx)) { best = ov; bestIdx = oi; }
    }
    if (lane == 0) { sVal[wid] = best; sIdx[wid] = bestIdx; }
    __syncthreads();

    // ---- cross-wave reduction in wave 0 (exactly 32 partials)
    if (wid == 0) {
      best = sVal[lane];
      bestIdx = sIdx[lane];
#pragma unroll
      for (int off = 16; off > 0; off >>= 1) {
        const float ov = __shfl_down(best, off, 32);
        const int   oi = __shfl_down(bestIdx, off, 32);
        if (ov > best || (ov == best && oi < bestIdx)) { best = ov; bestIdx = oi; }
      }
      if (lane == 0) {
        const float* sp = xb + (size_t)bestIdx * 3;
        const float nx = sp[0], ny = sp[1], nz = sp[2];
        sPoint[0] = nx; sPoint[1] = ny; sPoint[2] = nz;
        float* yo = yb + (size_t)s * 3;
        yo[0] = nx; yo[1] = ny; yo[2] = nz;
      }
    }
    __syncthreads();

    px = sPoint[0]; py = sPoint[1]; pz = sPoint[2];
  }
}

extern "C" void kernel_launch(void* const* d_in, const int* in_sizes, int n_in,
                              void* d_out, int out_size, void* d_ws, size_t ws_size,
                              hipStream_t stream) {
  const float* x = (const float*)d_in[0];
  float* y = (float*)d_out;

  const int M = 1024;                               // N_SAMPLES
  const int B = out_size / (3 * M);                 // 16
  const long long total = in_sizes[0];              // B*N*3
  const int N = (int)(total / ((long long)B * 3));  // 131072

  float* temp = (float*)d_ws;                       // B*N floats (8 MB)

  fps_kernel<<<B, TPB, 0, stream>>>(x, y, temp, N, M);
}